// GraphConvolutionNetwork_49881750175959
// MI455X (gfx1250) — compile-verified
//
#include <hip/hip_runtime.h>

typedef __attribute__((ext_vector_type(2))) float v2f;
typedef __attribute__((ext_vector_type(8))) float v8f;

#define N_NODES_C   100000
#define N_EDGES_C   1600000
#define IN_FEATS_C  128
#define HIDDEN_C    64
#define NCLASS_C    16

// ---------------- utility: zero a float buffer (graph-capture-safe) ----------
__global__ void k_zero(float* __restrict__ p, long n) {
    long i = (long)blockIdx.x * blockDim.x + threadIdx.x;
    long stride = (long)gridDim.x * blockDim.x;
    for (; i < n; i += stride) p[i] = 0.0f;
}

// ---------------- degree accumulation (float counts are exact < 2^24) --------
__global__ void k_degree(const int* __restrict__ src, const int* __restrict__ dst,
                         float* __restrict__ deg_out, float* __restrict__ deg_in,
                         int n_edges) {
    int e = blockIdx.x * blockDim.x + threadIdx.x;
    if (e < n_edges) {
        unsafeAtomicAdd(&deg_out[src[e]], 1.0f);
        unsafeAtomicAdd(&deg_in[dst[e]], 1.0f);
    }
}

// ---------------- norm = rsqrt(max(deg, 1)) in place -------------------------
__global__ void k_rsqrt_clip(float* __restrict__ buf, int n) {
    int i = blockIdx.x * blockDim.x + threadIdx.x;
    if (i < n) {
        float v = buf[i];
        buf[i] = rsqrtf(v < 1.0f ? 1.0f : v);
    }
}

// ---------------- H = (X .* norm_row) @ W via V_WMMA_F32_16X16X4_F32 ---------
// One wave computes a 16-row x NOUT output tile. K stepped by 4.
// A layout (16x4 f32): lanes 0-15 -> K=k,k+1 ; lanes 16-31 -> K=k+2,k+3 (M = lane%16)
// B layout (4x16 f32): VGPR0 -> row k (lanes 0-15) / k+2 (lanes 16-31); VGPR1 -> k+1 / k+3
// C/D layout: VGPR j -> M = j (lanes 0-15) or j+8 (lanes 16-31), N = lane%16
template <int K, int NOUT>
__global__ __launch_bounds__(256) void k_gemm_norm(const float* __restrict__ X,
                                                   const float* __restrict__ nrm,
                                                   const float* __restrict__ W,
                                                   float* __restrict__ H,
                                                   int n_tiles) {
    const int wave = threadIdx.x >> 5;
    const int lane = threadIdx.x & 31;
    const int tile = blockIdx.x * 8 + wave;
    if (tile >= n_tiles) return;              // wave-uniform: EXEC stays all-ones
    const int half = lane >> 4;               // 0 or 1
    const int l16  = lane & 15;
    const int row  = tile * 16 + l16;
    const float s  = nrm[row];
    const float* __restrict__ xr = X + (long)row * K;

    v8f acc[NOUT / 16] = {};
    for (int k = 0; k < K; k += 4) {
        const int kk = k + half * 2;
        v2f a;
        a.x = xr[kk] * s;
        a.y = xr[kk + 1] * s;
#pragma unroll
        for (int t = 0; t < NOUT / 16; ++t) {
            v2f b;
            b.x = W[(long)kk * NOUT + t * 16 + l16];
            b.y = W[(long)(kk + 1) * NOUT + t * 16 + l16];
            acc[t] = __builtin_amdgcn_wmma_f32_16x16x4_f32(
                false, a, false, b, (short)0, acc[t], false, false);
        }
    }
#pragma unroll
    for (int t = 0; t < NOUT / 16; ++t) {
#pragma unroll
        for (int j = 0; j < 8; ++j) {
            const int r = tile * 16 + half * 8 + j;
            H[(long)r * NOUT + t * 16 + l16] = acc[t][j];
        }
    }
}

// ---------------- edge scatter-add: agg[dst] += h[src] (F floats per node) ---
template <int F>
__global__ void k_scatter(const float* __restrict__ h, const int* __restrict__ src,
                          const int* __restrict__ dst, float* __restrict__ agg,
                          int n_edges) {
    const int cpe = F / 4;                    // float4 chunks per edge
    long idx = (long)blockIdx.x * blockDim.x + threadIdx.x;
    if (idx >= (long)n_edges * cpe) return;
    const int e = (int)(idx / cpe);
    const int c = (int)(idx % cpe);
    const float4 v = *(const float4*)(h + (long)src[e] * F + c * 4);
    float* ap = agg + (long)dst[e] * F + c * 4;
    unsafeAtomicAdd(ap + 0, v.x);
    unsafeAtomicAdd(ap + 1, v.y);
    unsafeAtomicAdd(ap + 2, v.z);
    unsafeAtomicAdd(ap + 3, v.w);
}

// ---------------- out = relu(agg * norm_dst + b) -----------------------------
template <int F>
__global__ void k_finalize_relu(const float* __restrict__ agg, const float* __restrict__ nd,
                                const float* __restrict__ b, float* __restrict__ out,
                                int n_nodes) {
    long idx = (long)blockIdx.x * blockDim.x + threadIdx.x;
    if (idx >= (long)n_nodes * F) return;
    const int i = (int)(idx / F);
    const int f = (int)(idx % F);
    float v = agg[idx] * nd[i] + b[f];
    out[idx] = v > 0.0f ? v : 0.0f;
}

// ---------------- out = softmax(agg * norm_dst + b) over 16 classes ----------
__global__ void k_softmax16(const float* __restrict__ agg, const float* __restrict__ nd,
                            const float* __restrict__ b, float* __restrict__ out,
                            int n_nodes) {
    int i = blockIdx.x * blockDim.x + threadIdx.x;
    if (i >= n_nodes) return;
    const float s = nd[i];
    float v[16];
    float mx = -3.402823466e38f;
#pragma unroll
    for (int c = 0; c < 16; ++c) {
        v[c] = agg[(long)i * 16 + c] * s + b[c];
        mx = fmaxf(mx, v[c]);
    }
    float sum = 0.0f;
#pragma unroll
    for (int c = 0; c < 16; ++c) {
        v[c] = expf(v[c] - mx);
        sum += v[c];
    }
    const float inv = 1.0f / sum;
#pragma unroll
    for (int c = 0; c < 16; ++c) out[(long)i * 16 + c] = v[c] * inv;
}

extern "C" void kernel_launch(void* const* d_in, const int* in_sizes, int n_in,
                              void* d_out, int out_size, void* d_ws, size_t ws_size,
                              hipStream_t stream) {
    const float* x  = (const float*)d_in[0];
    const int*   src = (const int*)d_in[1];
    const int*   dst = (const int*)d_in[2];
    const float* W1 = (const float*)d_in[3];
    const float* b1 = (const float*)d_in[4];
    const float* W2 = (const float*)d_in[5];
    const float* b2 = (const float*)d_in[6];
    float* out = (float*)d_out;

    // workspace layout (floats): [norm_src N][norm_dst N][bufA N*64][bufB N*64]
    float* ws       = (float*)d_ws;
    float* norm_src = ws;                                  // N
    float* norm_dst = ws + N_NODES_C;                      // N
    float* bufA     = ws + 2L * N_NODES_C;                 // N*64 : h1 = X@W1, then relu'd h1
    float* bufB     = bufA + (long)N_NODES_C * HIDDEN_C;   // N*64 : agg1, then reused:
    float* h2       = bufB;                                //   N*16 : h1@W2
    float* agg2     = bufB + (long)N_NODES_C * NCLASS_C;   //   N*16 : layer-2 aggregation

    const int B = 256;
    const int tiles = N_NODES_C / 16;                      // 6250, exact
    const int gemm_blocks = (tiles + 7) / 8;

    // degrees -> norms (shared by both layers)
    k_zero<<<512, B, 0, stream>>>(norm_src, 2L * N_NODES_C);
    k_degree<<<(N_EDGES_C + B - 1) / B, B, 0, stream>>>(src, dst, norm_src, norm_dst, N_EDGES_C);
    k_rsqrt_clip<<<(2 * N_NODES_C + B - 1) / B, B, 0, stream>>>(norm_src, 2 * N_NODES_C);

    // layer 1
    k_zero<<<2048, B, 0, stream>>>(bufB, (long)N_NODES_C * HIDDEN_C);
    k_gemm_norm<IN_FEATS_C, HIDDEN_C><<<gemm_blocks, B, 0, stream>>>(x, norm_src, W1, bufA, tiles);
    {
        long t = (long)N_EDGES_C * (HIDDEN_C / 4);
        k_scatter<HIDDEN_C><<<(int)((t + B - 1) / B), B, 0, stream>>>(bufA, src, dst, bufB, N_EDGES_C);
    }
    {
        long t = (long)N_NODES_C * HIDDEN_C;
        k_finalize_relu<HIDDEN_C><<<(int)((t + B - 1) / B), B, 0, stream>>>(bufB, norm_dst, b1, bufA, N_NODES_C);
    }

    // layer 2 (agg1 in bufB is dead now; reuse bufB for h2 + agg2)
    k_gemm_norm<HIDDEN_C, NCLASS_C><<<gemm_blocks, B, 0, stream>>>(bufA, norm_src, W2, h2, tiles);
    k_zero<<<512, B, 0, stream>>>(agg2, (long)N_NODES_C * NCLASS_C);
    {
        long t = (long)N_EDGES_C * (NCLASS_C / 4);
        k_scatter<NCLASS_C><<<(int)((t + B - 1) / B), B, 0, stream>>>(h2, src, dst, agg2, N_EDGES_C);
    }
    k_softmax16<<<(N_NODES_C + B - 1) / B, B, 0, stream>>>(agg2, norm_dst, b2, out, N_NODES_C);
}